// MultiGCN_30897994727996
// MI455X (gfx1250) — compile-verified
//
#include <hip/hip_runtime.h>

// ---------------------------------------------------------------------------
// MultiGCN forward for MI455X (gfx1250, wave32).
// Scatter-bound workload (~10.4 GB irregular traffic @ 23.3 TB/s HBM / L2).
// Dense h@W done with V_WMMA_F32_16X16X4_F32 (exact fp32, matrix pipe).
// ---------------------------------------------------------------------------

#define BGRAPH 64
#define DFEAT  128
#define FCDIM  768
#define EPSV   1e-5f
#define SLOPE  0.01f

typedef __attribute__((ext_vector_type(2))) float v2f;
typedef __attribute__((ext_vector_type(8))) float v8f;

static __device__ __forceinline__ float leakyf(float v) {
  return v > 0.f ? v : SLOPE * v;
}

// x0 = concat(x[N,127], pLDDT[N,1]) -> [N,128]
__global__ __launch_bounds__(256) void k_concat(const float* __restrict__ x,
    const float* __restrict__ pl, float* __restrict__ x0, int n) {
  long long i = (long long)blockIdx.x * blockDim.x + threadIdx.x;
  if (i >= (long long)n * DFEAT) return;
  int node = (int)(i >> 7);
  int f = (int)(i & 127);
  x0[i] = (f < 127) ? x[(long long)node * 127 + f] : pl[node];
}

// per-graph contiguous ranges (batch is sorted)
__global__ void k_bounds_init(int* __restrict__ gs, int* __restrict__ ge, int n) {
  int i = threadIdx.x;
  if (i < BGRAPH) { gs[i] = n; ge[i] = 0; }
}
__global__ __launch_bounds__(256) void k_bounds(const int* __restrict__ batch,
    int* __restrict__ gs, int* __restrict__ ge, int n) {
  int i = blockIdx.x * blockDim.x + threadIdx.x;
  if (i >= n) return;
  int b = batch[i];
  atomicMin(&gs[b], i);
  atomicMax(&ge[b], i + 1);
}

// degree with self-loop: deg starts at 1.0 per node
__global__ __launch_bounds__(256) void k_deg_init(float* __restrict__ deg, int n) {
  int i = blockIdx.x * blockDim.x + threadIdx.x;
  if (i < n) deg[i] = 1.0f;
}
__global__ __launch_bounds__(256) void k_deg_acc(const int* __restrict__ dst,
    float* __restrict__ deg, int ne) {
  int e = blockIdx.x * blockDim.x + threadIdx.x;
  if (e < ne) atomicAdd(&deg[dst[e]], 1.0f);
}
__global__ __launch_bounds__(256) void k_deg_fin(float* __restrict__ dis, int n) {
  int i = blockIdx.x * blockDim.x + threadIdx.x;
  if (i < n) dis[i] = rsqrtf(dis[i]);   // deg >= 1 always (self loop)
}

// GraphNorm stats: one block per graph, 128 threads (one per feature).
// Coalesced 512B row reads; Σh and Σh² in one pass.
__global__ __launch_bounds__(128) void k_gn_stats(const float* __restrict__ h,
    const int* __restrict__ gs, const int* __restrict__ ge,
    float* __restrict__ meanv, float* __restrict__ eh2v) {
  int b = blockIdx.x;
  int f = threadIdx.x;
  int s = gs[b], e = ge[b];
  float sum = 0.f, sq = 0.f;
  for (int nIdx = s; nIdx < e; ++nIdx) {
    float v = h[(long long)nIdx * DFEAT + f];
    sum += v; sq += v * v;
  }
  int c = e - s; if (c < 1) c = 1;
  float inv = 1.0f / (float)c;
  meanv[b * DFEAT + f] = sum * inv;
  eh2v[b * DFEAT + f] = sq * inv;
}

// out = w * (h - mean*ms) * rsqrt(var + eps) + b ; var = E[h^2] - 2c*m + c^2
__global__ __launch_bounds__(256) void k_gn_apply(const float* __restrict__ h,
    const int* __restrict__ batch, const float* __restrict__ meanv,
    const float* __restrict__ eh2v, const float* __restrict__ w,
    const float* __restrict__ bp, const float* __restrict__ ms,
    float* __restrict__ out, int n) {
  long long i = (long long)blockIdx.x * blockDim.x + threadIdx.x;
  if (i >= (long long)n * DFEAT) return;
  int node = (int)(i >> 7);
  int f = (int)(i & 127);
  int b = batch[node];
  float m = meanv[b * DFEAT + f];
  float c = m * ms[f];
  float var = eh2v[b * DFEAT + f] - 2.f * c * m + c * c;
  if (var < 0.f) var = 0.f;
  float sub = h[i] - c;
  out[i] = w[f] * sub * rsqrtf(var + EPSV) + bp[f];
}

// hw = A[N,128] @ W[128,128] via V_WMMA_F32_16X16X4_F32.
// Block = 256 threads = 8 waves; wave w -> column tile w; block -> 16-row tile.
// A frag (16x4 f32): v[0] lanes0-15 K+0 / lanes16-31 K+2; v[1] K+1 / K+3.
// B frag (4x16 f32): same striping over W rows. C/D: 8 VGPRs, M=r (lo) / r+8 (hi).
__global__ __launch_bounds__(256) void k_gemm_wmma(const float* __restrict__ A,
    const float* __restrict__ W, float* __restrict__ C, int nrows) {
  int wave = threadIdx.x >> 5;
  int lane = threadIdx.x & 31;
  int half = lane >> 4;
  int l16 = lane & 15;
  int row0 = blockIdx.x * 16;
  int col0 = wave * 16;
  int arow = row0 + l16;
  if (arow >= nrows) arow = nrows - 1;        // keep EXEC all-1 for WMMA
  const float* Ap = A + (long long)arow * DFEAT + half * 2;
  const float* Wp = W + (long long)(half * 2) * DFEAT + col0 + l16;
  v8f c = {0.f, 0.f, 0.f, 0.f, 0.f, 0.f, 0.f, 0.f};
  for (int k = 0; k < DFEAT; k += 4) {
    v2f a, b;
    a.x = Ap[k];
    a.y = Ap[k + 1];
    b.x = Wp[(long long)k * DFEAT];
    b.y = Wp[(long long)(k + 1) * DFEAT];
    c = __builtin_amdgcn_wmma_f32_16x16x4_f32(false, a, false, b, (short)0, c,
                                              false, false);
  }
#pragma unroll
  for (int r = 0; r < 8; ++r) {
    int row = row0 + half * 8 + r;
    if (row < nrows) C[(long long)row * DFEAT + col0 + l16] = c[r];
  }
}

// agg = dis[i]^2 * hw[i]  (fused self-loop contribution; fully initializes agg)
__global__ __launch_bounds__(256) void k_selfloop(const float* __restrict__ hw,
    const float* __restrict__ dis, float* __restrict__ agg, int n) {
  long long i = (long long)blockIdx.x * blockDim.x + threadIdx.x;
  if (i >= (long long)n * DFEAT) return;
  int node = (int)(i >> 7);
  float w = dis[node] * dis[node];
  agg[i] = w * hw[i];
}

// edge scatter: one wave32 per edge (wave-uniform e), float4 gather + 4 atomics
__global__ __launch_bounds__(256) void k_scatter(const int* __restrict__ src,
    const int* __restrict__ dst, const float* __restrict__ dis,
    const float* __restrict__ hw, float* __restrict__ agg, int ne) {
  long long t = (long long)blockIdx.x * blockDim.x + threadIdx.x;
  int e = (int)(t >> 5);
  int lane = (int)(t & 31);
  if (e >= ne) return;
  int s = src[e];
  int d = dst[e];
  float w = dis[s] * dis[d];
  const float4* hp = (const float4*)(hw + (long long)s * DFEAT);
  float4 v = hp[lane];
  float* ap = agg + (long long)d * DFEAT + lane * 4;
  atomicAdd(ap + 0, w * v.x);
  atomicAdd(ap + 1, w * v.y);
  atomicAdd(ap + 2, w * v.z);
  atomicAdd(ap + 3, w * v.w);
}

// h = leaky(agg + bias)
__global__ __launch_bounds__(256) void k_bias_leaky(const float* __restrict__ agg,
    const float* __restrict__ bias, float* __restrict__ h, int n) {
  long long i = (long long)blockIdx.x * blockDim.x + threadIdx.x;
  if (i >= (long long)n * DFEAT) return;
  int f = (int)(i & 127);
  h[i] = leakyf(agg[i] + bias[f]);
}

// per-graph mean pool of one layer's output into pooled[:, col_off:col_off+128]
__global__ __launch_bounds__(128) void k_pool(const float* __restrict__ h,
    const int* __restrict__ gs, const int* __restrict__ ge,
    float* __restrict__ pooled, int col_off) {
  int b = blockIdx.x;
  int f = threadIdx.x;
  int s = gs[b], e = ge[b];
  float sum = 0.f;
  for (int nIdx = s; nIdx < e; ++nIdx) sum += h[(long long)nIdx * DFEAT + f];
  int c = e - s; if (c < 1) c = 1;
  pooled[(long long)b * FCDIM + col_off + f] = sum / (float)c;
}

// training-mode BatchNorm over rows (biased variance)
__global__ __launch_bounds__(256) void k_bn(const float* __restrict__ in,
    const float* __restrict__ g, const float* __restrict__ bb,
    float* __restrict__ out, int rows, int cols) {
  int c = blockIdx.x * blockDim.x + threadIdx.x;
  if (c >= cols) return;
  float s = 0.f, sq = 0.f;
  for (int r = 0; r < rows; ++r) {
    float v = in[(long long)r * cols + c];
    s += v; sq += v * v;
  }
  float m = s / rows;
  float var = sq / rows - m * m;
  if (var < 0.f) var = 0.f;
  float inv = rsqrtf(var + EPSV);
  float gg = g[c], bc = bb[c];
  for (int r = 0; r < rows; ++r) {
    float v = in[(long long)r * cols + c];
    out[(long long)r * cols + c] = gg * (v - m) * inv + bc;
  }
}

// fc1: [64,768] @ [768,128] + b, leaky
__global__ __launch_bounds__(128) void k_fc1(const float* __restrict__ in,
    const float* __restrict__ W, const float* __restrict__ bias,
    float* __restrict__ out) {
  int r = blockIdx.x;
  int c = threadIdx.x;
  float s = bias[c];
  const float* ip = in + (long long)r * FCDIM;
  for (int k = 0; k < FCDIM; ++k) s += ip[k] * W[(long long)k * DFEAT + c];
  out[(long long)r * DFEAT + c] = leakyf(s);
}

// fc2: [64,128] @ [128,1] + b
__global__ __launch_bounds__(64) void k_fc2(const float* __restrict__ in,
    const float* __restrict__ W, const float* __restrict__ bias,
    float* __restrict__ out) {
  int r = threadIdx.x;
  if (r >= BGRAPH) return;
  float s = 0.f;
  const float* ip = in + (long long)r * DFEAT;
  for (int k = 0; k < DFEAT; ++k) s += ip[k] * W[k];
  out[r] = s + bias[0];
}

// ---------------------------------------------------------------------------
extern "C" void kernel_launch(void* const* d_in, const int* in_sizes, int n_in,
                              void* d_out, int out_size, void* d_ws, size_t ws_size,
                              hipStream_t stream) {
  (void)n_in; (void)out_size; (void)ws_size;

  // ---- input map (setup_inputs insertion order, params pytree in insertion
  //      order: branches[2][3]{gn_w,gn_b,gn_ms,W,b}, bn1_g,bn1_b,fc1_W,fc1_b,
  //      bn2_g,bn2_b,fc2_W,fc2_b) ----
  const float* x     = (const float*)d_in[0];   // [N,127]
  const float* pLDDT = (const float*)d_in[1];   // [N]
  const int N = in_sizes[1];
  const int PARAM0 = 2;                          // first branch param
  const int TAIL0 = PARAM0 + 2 * 3 * 5;          // = 32: bn1_g ...
  const float* bn1_g = (const float*)d_in[TAIL0 + 0];
  const float* bn1_b = (const float*)d_in[TAIL0 + 1];
  const float* fc1_W = (const float*)d_in[TAIL0 + 2];
  const float* fc1_b = (const float*)d_in[TAIL0 + 3];
  const float* bn2_g = (const float*)d_in[TAIL0 + 4];
  const float* bn2_b = (const float*)d_in[TAIL0 + 5];
  const float* fc2_W = (const float*)d_in[TAIL0 + 6];
  const float* fc2_b = (const float*)d_in[TAIL0 + 7];
  const int* ei_c  = (const int*)d_in[TAIL0 + 8];   // [2,E]
  const int* ei_b  = (const int*)d_in[TAIL0 + 9];   // [2,E]
  const int* batch = (const int*)d_in[TAIL0 + 10];  // [N]
  const int E = in_sizes[TAIL0 + 8] / 2;

  // ---- workspace layout (floats) ----
  float* ws = (float*)d_ws;
  const long long NF = (long long)N * DFEAT;
  float* x0  = ws;               // [N,128]
  float* hA  = x0 + NF;          // current layer output
  float* hB  = hA + NF;          // graphnorm out, then reused as agg
  float* hC  = hB + NF;          // h @ W
  float* dis = hC + NF;          // [N]
  float* meanv  = dis + N;       // [B,128]
  float* eh2v   = meanv + BGRAPH * DFEAT;
  float* pooled = eh2v + BGRAPH * DFEAT;          // [B,768]
  float* bn1o   = pooled + BGRAPH * FCDIM;        // [B,768]
  float* fc1o   = bn1o + BGRAPH * FCDIM;          // [B,128]
  float* bn2o   = fc1o + BGRAPH * DFEAT;          // [B,128]
  int* gstart = (int*)(bn2o + BGRAPH * DFEAT);    // [B]
  int* gend   = gstart + BGRAPH;                  // [B]

  const int T = 256;
  const long long tot = NF;
  const int gElem = (int)((tot + T - 1) / T);       // N*128 elementwise grid
  const int gNode = (N + T - 1) / T;
  const int gEdge = (E + T - 1) / T;
  const int gScat = (int)(((long long)E * 32 + T - 1) / T);
  const int gTile = (N + 15) / 16;                  // 16-row WMMA tiles

  // ---- shared preprocessing ----
  k_concat<<<gElem, T, 0, stream>>>(x, pLDDT, x0, N);
  k_bounds_init<<<1, BGRAPH, 0, stream>>>(gstart, gend, N);
  k_bounds<<<gNode, T, 0, stream>>>(batch, gstart, gend, N);

  // ---- two GCN branches ----
  for (int br = 0; br < 2; ++br) {
    const int* ei  = (br == 0) ? ei_c : ei_b;
    const int* src = ei;
    const int* dst = ei + E;

    k_deg_init<<<gNode, T, 0, stream>>>(dis, N);
    k_deg_acc<<<gEdge, T, 0, stream>>>(dst, dis, E);
    k_deg_fin<<<gNode, T, 0, stream>>>(dis, N);

    const float* hin = x0;
    for (int li = 0; li < 3; ++li) {
      const int pb = PARAM0 + (br * 3 + li) * 5;
      const float* gn_w  = (const float*)d_in[pb + 0];
      const float* gn_b  = (const float*)d_in[pb + 1];
      const float* gn_ms = (const float*)d_in[pb + 2];
      const float* Wl    = (const float*)d_in[pb + 3];
      const float* bl    = (const float*)d_in[pb + 4];

      k_gn_stats<<<BGRAPH, DFEAT, 0, stream>>>(hin, gstart, gend, meanv, eh2v);
      k_gn_apply<<<gElem, T, 0, stream>>>(hin, batch, meanv, eh2v,
                                          gn_w, gn_b, gn_ms, hB, N);
      k_gemm_wmma<<<gTile, T, 0, stream>>>(hB, Wl, hC, N);
      k_selfloop<<<gElem, T, 0, stream>>>(hC, dis, hB, N);   // agg := hB
      k_scatter<<<gScat, T, 0, stream>>>(src, dst, dis, hC, hB, E);
      k_bias_leaky<<<gElem, T, 0, stream>>>(hB, bl, hA, N);
      k_pool<<<BGRAPH, DFEAT, 0, stream>>>(hA, gstart, gend, pooled,
                                           (br * 3 + li) * DFEAT);
      hin = hA;
    }
  }

  // ---- head MLP on [64,768] ----
  k_bn<<<(FCDIM + T - 1) / T, T, 0, stream>>>(pooled, bn1_g, bn1_b, bn1o,
                                              BGRAPH, FCDIM);
  k_fc1<<<BGRAPH, DFEAT, 0, stream>>>(bn1o, fc1_W, fc1_b, fc1o);
  k_bn<<<1, DFEAT, 0, stream>>>(fc1o, bn2_g, bn2_b, bn2o, BGRAPH, DFEAT);
  k_fc2<<<1, BGRAPH, 0, stream>>>(bn2o, fc2_W, fc2_b, (float*)d_out);
}